// MatrixNet_31834297598631
// MI455X (gfx1250) — compile-verified
//
#include <hip/hip_runtime.h>
#include <math.h>

#define KN 1024      // variable nodes
#define NN 512       // check nodes
#define EE 3072      // edges (KN*3)
#define BB 2048      // batch
#define PAD 32       // padded check degree (Poisson(6) tail: P(deg>32) ~ 1e-16)
#define LWS 33       // LDS row stride (bank-conflict pad)
#define CLIPV 10.0f

typedef float v2f __attribute__((ext_vector_type(2)));
typedef float v8f __attribute__((ext_vector_type(8)));

// ---------------- setup kernels (cheap, once per call, deterministic) -------

__global__ __launch_bounds__(256)
void zero_kernel(float* __restrict__ p, int n) {
    int i = blockIdx.x * 256 + threadIdx.x;
    if (i < n) p[i] = 0.0f;
}

// xT[k, b] = x[b, k]   (tiled transpose so WMMA A-gathers are coalesced)
__global__ __launch_bounds__(256)
void transpose_kernel(const float* __restrict__ x, float* __restrict__ xT) {
    __shared__ float tile[32][33];
    const int k0 = blockIdx.x * 32;
    const int b0 = blockIdx.y * 32;
    const int tx = threadIdx.x & 31, ty = threadIdx.x >> 5; // 8 rows of 32
    for (int i = ty; i < 32; i += 8)
        tile[i][tx] = x[(size_t)(b0 + i) * KN + k0 + tx];
    __syncthreads();
    for (int i = ty; i < 32; i += 8)
        xT[(size_t)(k0 + i) * BB + b0 + tx] = tile[tx][i];
}

// chk[e] = argmax_n out_mask[e, n]  (exactly one 1.0 per row)
__global__ __launch_bounds__(256)
void chk_kernel(const float* __restrict__ out_mask, int* __restrict__ chk) {
    int e = blockIdx.x * 256 + threadIdx.x;
    if (e < EE) {
        int c = 0;
        const float* row = out_mask + (size_t)e * NN;
        for (int n = 0; n < NN; ++n)
            if (row[n] > 0.5f) c = n;
        chk[e] = c;
    }
}

// perm[c][s] = s-th edge (ascending) attached to check c, -1 past degree.
// Deterministic (no atomics) so fp summation order is replay-stable.
__global__ __launch_bounds__(256)
void lists_kernel(const int* __restrict__ chk, int* __restrict__ perm) {
    int c = blockIdx.x * 256 + threadIdx.x;
    if (c < NN) {
        int d = 0;
        for (int e = 0; e < EE; ++e)
            if (chk[e] == c && d < PAD) perm[c * PAD + d++] = e;
        for (; d < PAD; ++d) perm[c * PAD + d] = -1;
    }
}

// Gather the (o2e_mask * we) nonzeros into dense 32x32 per-check blocks:
// Wblk[l][c][s][t] = we_l[perm[c][s], perm[c][t]]  (s!=t, both valid)
__global__ __launch_bounds__(256)
void wblk_kernel(const int* __restrict__ perm,
                 const float* __restrict__ we1, const float* __restrict__ we2,
                 const float* __restrict__ we3, float* __restrict__ wblk) {
    int idx = blockIdx.x * 256 + threadIdx.x;      // 0 .. NN*PAD*PAD
    int l = blockIdx.y;
    const float* we = (l == 0) ? we1 : (l == 1) ? we2 : we3;
    int c = idx >> 10;
    int s = (idx >> 5) & 31;
    int t = idx & 31;
    int es = perm[c * PAD + s], et = perm[c * PAD + t];
    float v = 0.0f;
    if (es >= 0 && et >= 0 && s != t) v = we[(size_t)es * EE + et];
    wblk[(size_t)l * NN * PAD * PAD + idx] = v;
}

// ---------------- odd layer: block-diagonal WMMA f32 GEMM + tanh -----------
// odd[b,j] = tanh(0.5*clip(lv[b,j]*wv[j] + sum_s e_in[b,perm[c,s]]*W[s,t], +-10))
// One wave per (check, 16-batch tile); 8 waves/block share the LDS W tile.
// All slot-dependent address math (validity, lv /3 mapping) is resolved ONCE
// per block into an LDS pointer table; invalid slots alias a zero row, so the
// hot loop is branch-free: 2 coalesced global loads + LDS loads + 2 WMMAs.
__global__ __launch_bounds__(256)
void odd_wmma_kernel(const float* __restrict__ einT,  // (E,B), ignored if lv_mode
                     const float* __restrict__ xT,    // (K,B)
                     const float* __restrict__ wv,    // (E)
                     const float* __restrict__ wblk,  // (NN,32,32) for this layer
                     const int*   __restrict__ perm,  // (NN,32)
                     const float* __restrict__ zrow,  // BB zeros
                     float* __restrict__ outT,        // (E,B)
                     int lv_mode) {
    __shared__ float        lw[PAD * LWS];
    __shared__ const float* lrow[PAD];   // A-matrix row base per slot
    __shared__ const float* lx[PAD];     // lv (channel) row base per slot
    __shared__ float        lwv[PAD];
    __shared__ int          lperm[PAD];

    const int c   = blockIdx.x;
    const int tid = threadIdx.x;
    for (int i = tid; i < PAD * PAD; i += 256)
        lw[(i >> 5) * LWS + (i & 31)] = wblk[(size_t)c * PAD * PAD + i];
    if (tid < PAD) {
        int e = perm[c * PAD + tid];
        lperm[tid] = e;
        lwv[tid]   = (e >= 0) ? wv[e] : 0.0f;
        const float* pa = zrow;
        const float* px = zrow;
        if (e >= 0) {
            px = xT + (size_t)(e / 3) * BB;
            pa = lv_mode ? px : (einT + (size_t)e * BB);
        }
        lrow[tid] = pa;
        lx[tid]   = px;
    }
    __syncthreads();

    const int wave = tid >> 5;
    const int lane = tid & 31;
    const int half = lane >> 4;   // K-half (A/B frags) and M-half (C/D frag)
    const int lm   = lane & 15;   // M (A) / N (B,D) within half
    const int b0   = (blockIdx.y * 8 + wave) * 16;

    v8f acc0 = {}; v8f acc1 = {};
    for (int kk = 0; kk < 8; ++kk) {
        const int s0 = kk * 4 + half * 2;            // this lane's two K slots
        // A fragment: rows = batch (coalesced 64B per half), cols = slots
        v2f a;
        a.x = lrow[s0][b0 + lm];
        a.y = lrow[s0 + 1][b0 + lm];
        // B fragments: weight block from LDS, two 16-col halves
        v2f bw0, bw1;
        bw0.x = lw[s0 * LWS + lm];       bw0.y = lw[(s0 + 1) * LWS + lm];
        bw1.x = lw[s0 * LWS + 16 + lm];  bw1.y = lw[(s0 + 1) * LWS + 16 + lm];
        acc0 = __builtin_amdgcn_wmma_f32_16x16x4_f32(false, a, false, bw0,
                                                     (short)0, acc0, false, false);
        acc1 = __builtin_amdgcn_wmma_f32_16x16x4_f32(false, a, false, bw1,
                                                     (short)0, acc1, false, false);
    }

    // epilogue: + lv*wv, clip, tanh(0.5*), scatter to (E,B)
    for (int tcol = 0; tcol < 2; ++tcol) {
        v8f acc = tcol ? acc1 : acc0;
        int t = tcol * 16 + lm;
        int j = lperm[t];
        if (j >= 0) {
            float wvj = lwv[t];
            const float* xrow = lx[t] + b0;
            float* orow = outT + (size_t)j * BB + b0;
            for (int r = 0; r < 8; ++r) {
                int m = half * 8 + r;                 // D-matrix M mapping
                float val = acc[r] + xrow[m] * wvj;
                val = fminf(fmaxf(val, -CLIPV), CLIPV);
                orow[m] = tanhf(0.5f * val);
            }
        }
    }
}

// ---------------- even layer: closed-form (e2o has exactly 2 nnz/col) ------
__device__ __forceinline__ float even_msg(float oa, float ob) {
    float mag = (fabsf(oa) + 1e-12f) * (fabsf(ob) + 1e-12f);
    float ev  = ((oa < 0.0f) != (ob < 0.0f)) ? -mag : mag;
    ev = fminf(fmaxf(ev, -1.0f + 1e-7f), 1.0f - 1e-7f);
    return logf((1.0f + ev) / (1.0f - ev));           // 2*artanh
}

__global__ __launch_bounds__(256)
void even_kernel(const float* __restrict__ oddT, float* __restrict__ evT) {
    int idx = blockIdx.x * 256 + threadIdx.x;   // over KN*BB
    int v = idx >> 11;                          // / BB
    int b = idx & (BB - 1);
    const float* irow = oddT + (size_t)(3 * v) * BB + b;
    float o0 = irow[0], o1 = irow[BB], o2 = irow[2 * BB];
    float* orow = evT + (size_t)(3 * v) * BB + b;
    orow[0]      = even_msg(o1, o2);
    orow[BB]     = even_msg(o0, o2);
    orow[2 * BB] = even_msg(o0, o1);
}

// ---------------- output: per-check sparse reduction + sigmoid -------------
__global__ __launch_bounds__(256)
void out_kernel(const float* __restrict__ oddT, const float* __restrict__ w_out,
                const int* __restrict__ perm, float* __restrict__ out) {
    int n = blockIdx.x;
    int b = blockIdx.y * 256 + threadIdx.x;
    float acc = 0.0f;
    for (int s = 0; s < PAD; ++s) {
        int e = perm[n * PAD + s];
        if (e >= 0)
            acc += oddT[(size_t)e * BB + b] * w_out[(size_t)e * NN + n];
    }
    out[(size_t)b * NN + n] = 1.0f / (1.0f + expf(-acc));
}

// ---------------------------------------------------------------------------
extern "C" void kernel_launch(void* const* d_in, const int* in_sizes, int n_in,
                              void* d_out, int out_size, void* d_ws, size_t ws_size,
                              hipStream_t stream) {
    const float* x     = (const float*)d_in[0];
    // d_in[1] in_mask, d_in[3] o2e_mask, d_in[4] e2o_mask: structure derived instead
    const float* omask = (const float*)d_in[2];
    const float* wv1   = (const float*)d_in[5];
    const float* we1   = (const float*)d_in[6];
    const float* wv2   = (const float*)d_in[7];
    const float* we2   = (const float*)d_in[8];
    const float* wv3   = (const float*)d_in[9];
    const float* we3   = (const float*)d_in[10];
    const float* w_out = (const float*)d_in[11];
    float* out = (float*)d_out;

    // workspace carve-up
    char* ws = (char*)d_ws;
    size_t off = 0;
    auto take = [&](size_t bytes) { void* p = ws + off; off += (bytes + 255) & ~(size_t)255; return p; };
    float* xT   = (float*)take((size_t)KN * BB * 4);            //  8 MB
    float* bufA = (float*)take((size_t)EE * BB * 4);            // 24 MB (oddT)
    float* bufB = (float*)take((size_t)EE * BB * 4);            // 24 MB (evT)
    float* wblk = (float*)take((size_t)3 * NN * PAD * PAD * 4); //  6 MB
    int*   perm = (int*)take((size_t)NN * PAD * 4);
    int*   chk  = (int*)take((size_t)EE * 4);
    float* zrow = (float*)take((size_t)BB * 4);                 // zero row
    (void)ws_size; (void)in_sizes; (void)n_in; (void)out_size;

    // setup
    zero_kernel<<<dim3(BB / 256), 256, 0, stream>>>(zrow, BB);
    transpose_kernel<<<dim3(KN / 32, BB / 32), 256, 0, stream>>>(x, xT);
    chk_kernel<<<dim3((EE + 255) / 256), 256, 0, stream>>>(omask, chk);
    lists_kernel<<<dim3((NN + 255) / 256), 256, 0, stream>>>(chk, perm);
    wblk_kernel<<<dim3(NN * PAD * PAD / 256, 3), 256, 0, stream>>>(perm, we1, we2, we3, wblk);

    const dim3 gOdd(NN, BB / (16 * 8));      // 512 checks x 16 batch-groups
    const dim3 gEven(KN * BB / 256);
    const dim3 gOut(NN, BB / 256);
    float* wb1 = wblk;
    float* wb2 = wblk + (size_t)NN * PAD * PAD;
    float* wb3 = wblk + (size_t)2 * NN * PAD * PAD;

    // iteration 1:  lv -> odd -> ev1
    odd_wmma_kernel<<<gOdd, 256, 0, stream>>>(xT, xT, wv1, wb1, perm, zrow, bufA, 1);
    even_kernel<<<gEven, 256, 0, stream>>>(bufA, bufB);
    // iteration 2:  ev1 -> odd -> ev2
    odd_wmma_kernel<<<gOdd, 256, 0, stream>>>(bufB, xT, wv2, wb2, perm, zrow, bufA, 0);
    even_kernel<<<gEven, 256, 0, stream>>>(bufA, bufB);
    // iteration 3:  ev2 -> odd
    odd_wmma_kernel<<<gOdd, 256, 0, stream>>>(bufB, xT, wv3, wb3, perm, zrow, bufA, 0);
    // output projection + sigmoid
    out_kernel<<<gOut, 256, 0, stream>>>(bufA, w_out, perm, out);
}